// DisentangledProductQuantizer_88244398063986
// MI455X (gfx1250) — compile-verified
//
#include <hip/hip_runtime.h>
#include <hip/hip_bf16.h>

// ---------------------------------------------------------------------------
// DisentangledProductQuantizer on gfx1250 (MI455X), fp32 WMMA path.
//   B=8, S=4096, E=1024, G=4, K=256, gd=256.  All matmuls via
//   v_wmma_f32_16x16x4_f32 (precision-correct fp32 matrix pipe).
//   Dual-accumulator tiling amortizes each operand fetch over 2 WMMAs.
// ---------------------------------------------------------------------------

typedef float v2f __attribute__((ext_vector_type(2)));
typedef float v8f __attribute__((ext_vector_type(8)));

#define WMMA_F32(a, b, c) \
    __builtin_amdgcn_wmma_f32_16x16x4_f32(false, (a), false, (b), (short)0, (c), false, false)

namespace pqcfg {
constexpr int Bdim = 8;
constexpr int Sdim = 4096;
constexpr int Edim = 1024;
constexpr int Gdim = 4;
constexpr int Kdim = 256;
constexpr int GD   = 256;                 // group dim
constexpr int ROWS = 16;                  // row tile (M) for K1
constexpr int ROWS2 = 32;                 // row tile for K2 (2 M-tiles/wave)
constexpr int NROW = Bdim * Sdim;         // 32768
constexpr int LDA  = GD + 4;              // padded LDS row stride (bank spread)
constexpr float BETA = 4.0f;
constexpr float LOSS_SCALE = BETA / (float)((long long)NROW * Edim);
}

// ---------------------------------------------------------------------------
// K0: per-code squared norms c2[g][k], and zero the loss accumulator.
// ---------------------------------------------------------------------------
__global__ void __launch_bounds__(256)
pq_prep_kernel(const float* __restrict__ codebooks,
               float* __restrict__ c2ws,
               float* __restrict__ loss) {
    using namespace pqcfg;
    int i = blockIdx.x * blockDim.x + threadIdx.x;
    if (i == 0) *loss = 0.0f;
    if (i < Gdim * Kdim) {
        const float* c = codebooks + (size_t)i * GD;
        float s = 0.0f;
        #pragma unroll 4
        for (int e = 0; e < GD; ++e) s += c[e] * c[e];
        c2ws[i] = s;
    }
}

// ---------------------------------------------------------------------------
// K1: fused  proj-GEMM -> p2 -> cross-GEMM -> argmin -> commitment.
// Grid: (NROW/16, G).  Block: 256 threads = 8 wave32.
// Each wave owns a 32-column slice: ONE k-loop, one A fragment feeding
// two WMMAs (two 16x16 N-tiles / two accumulators).
// ---------------------------------------------------------------------------
__global__ void __launch_bounds__(256)
pq_quantize_kernel(const float* __restrict__ features,
                   const float* __restrict__ proj_w,
                   const float* __restrict__ proj_b,
                   const float* __restrict__ codebooks,
                   const float* __restrict__ c2ws,
                   int*   __restrict__ idx_ws,
                   float* __restrict__ out_indices,
                   float* __restrict__ loss) {
    using namespace pqcfg;
    __shared__ float sA[ROWS * LDA];       // features tile (16x256, padded)
    __shared__ float sProj[ROWS * LDA];    // proj tile
    __shared__ float sP2[ROWS];
    __shared__ float sBestD[8][ROWS];
    __shared__ int   sBestI[8][ROWS];
    __shared__ int   sIdx[ROWS];
    __shared__ float sRed[8];

    const int tid  = threadIdx.x;
    const int lane = tid & 31;
    const int wave = tid >> 5;
    const int n    = lane & 15;            // fragment column / A row
    const int ksel = lane >> 4;            // K-pair select (A/B) & row half (C/D)
    const int g    = blockIdx.y;
    const int r0   = blockIdx.x * ROWS;

    // ---- load features tile: rows r0..r0+15, cols g*GD..g*GD+255 ----------
    for (int i = tid; i < ROWS * (GD / 4); i += 256) {
        int row = i / (GD / 4);
        int c4  = i % (GD / 4);
        float4 v = *(const float4*)(features + (size_t)(r0 + row) * Edim
                                             + (size_t)g * GD + c4 * 4);
        *(float4*)&sA[row * LDA + c4 * 4] = v;
    }
    __syncthreads();

    // ---- phase 2: proj tile = sA(16x256) @ proj_w[g](256x256) -------------
    const float* pw = proj_w + (size_t)g * GD * GD;
    {
        const int col0 = wave * 32 + n;
        const int col1 = col0 + 16;
        v8f acc0 = {}, acc1 = {};
        #pragma unroll 4
        for (int k0 = 0; k0 < GD; k0 += 4) {
            const int ka = k0 + 2 * ksel;
            v2f a;  a.x  = sA[n * LDA + ka];  a.y = sA[n * LDA + ka + 1];
            v2f b0; b0.x = pw[(size_t)ka * GD + col0];
                    b0.y = pw[(size_t)(ka + 1) * GD + col0];
            v2f b1; b1.x = pw[(size_t)ka * GD + col1];
                    b1.y = pw[(size_t)(ka + 1) * GD + col1];
            acc0 = WMMA_F32(a, b0, acc0);
            acc1 = WMMA_F32(a, b1, acc1);
        }
        const float bias0 = proj_b[g * GD + col0];
        const float bias1 = proj_b[g * GD + col1];
        #pragma unroll
        for (int r = 0; r < 8; ++r) {
            sProj[(r + 8 * ksel) * LDA + col0] = acc0[r] + bias0;
            sProj[(r + 8 * ksel) * LDA + col1] = acc1[r] + bias1;
        }
    }
    __syncthreads();

    // ---- p2 per row --------------------------------------------------------
    if (tid < ROWS) {
        float s = 0.0f;
        #pragma unroll 4
        for (int e = 0; e < GD; ++e) { float v = sProj[tid * LDA + e]; s += v * v; }
        sP2[tid] = s;
    }
    __syncthreads();

    // ---- phase 3: cross tile = sProj @ cb[g]^T, distances, local argmin ----
    const float* cbg = codebooks + (size_t)g * Kdim * GD;
    float bestD[8]; int bestI[8];
    #pragma unroll
    for (int r = 0; r < 8; ++r) { bestD[r] = 3.4e38f; bestI[r] = 0; }

    {
        const int code0 = wave * 32 + n;
        const int code1 = code0 + 16;
        v8f acc0 = {}, acc1 = {};
        #pragma unroll 4
        for (int k0 = 0; k0 < GD; k0 += 4) {
            const int ka = k0 + 2 * ksel;
            v2f a; a.x = sProj[n * LDA + ka]; a.y = sProj[n * LDA + ka + 1];
            const float2 bv0 = *(const float2*)(cbg + (size_t)code0 * GD + ka);
            const float2 bv1 = *(const float2*)(cbg + (size_t)code1 * GD + ka);
            v2f b0; b0.x = bv0.x; b0.y = bv0.y;
            v2f b1; b1.x = bv1.x; b1.y = bv1.y;
            acc0 = WMMA_F32(a, b0, acc0);
            acc1 = WMMA_F32(a, b1, acc1);
        }
        const float c2v0 = c2ws[g * Kdim + code0];
        const float c2v1 = c2ws[g * Kdim + code1];
        #pragma unroll
        for (int r = 0; r < 8; ++r) {
            const int m = r + 8 * ksel;
            const float d0 = sP2[m] - 2.0f * acc0[r] + c2v0;   // code0 < code1:
            if (d0 < bestD[r]) { bestD[r] = d0; bestI[r] = code0; }
            const float d1 = sP2[m] - 2.0f * acc1[r] + c2v1;   // strict < keeps
            if (d1 < bestD[r]) { bestD[r] = d1; bestI[r] = code1; }  // first-min
        }
    }
    // butterfly argmin across the 16 lanes that share each row half
    #pragma unroll
    for (int off = 1; off < 16; off <<= 1) {
        #pragma unroll
        for (int r = 0; r < 8; ++r) {
            float od = __shfl_xor(bestD[r], off, 32);
            int   oi = __shfl_xor(bestI[r], off, 32);
            if (od < bestD[r] || (od == bestD[r] && oi < bestI[r])) {
                bestD[r] = od; bestI[r] = oi;
            }
        }
    }
    if (n == 0) {
        #pragma unroll
        for (int r = 0; r < 8; ++r) {
            sBestD[wave][r + 8 * ksel] = bestD[r];
            sBestI[wave][r + 8 * ksel] = bestI[r];
        }
    }
    __syncthreads();

    if (tid < ROWS) {
        float bd = sBestD[0][tid]; int bi = sBestI[0][tid];
        #pragma unroll
        for (int w = 1; w < 8; ++w) {
            float d = sBestD[w][tid]; int i2 = sBestI[w][tid];
            if (d < bd || (d == bd && i2 < bi)) { bd = d; bi = i2; }
        }
        sIdx[tid] = bi;
        const size_t rr = (size_t)(r0 + tid);
        idx_ws[rr * Gdim + g]      = bi;
        out_indices[rr * Gdim + g] = (float)bi;   // int32 index, value-cast
    }
    __syncthreads();

    // ---- phase 4: commitment loss contribution -----------------------------
    float local = 0.0f;
    {
        const int m  = tid >> 4;
        const int e0 = tid & 15;
        const float* cq = cbg + (size_t)sIdx[m] * GD;
        #pragma unroll 4
        for (int j = 0; j < 16; ++j) {
            const int e = e0 + j * 16;
            const float diff = cq[e] - sProj[m * LDA + e];
            local += diff * diff;
        }
    }
    #pragma unroll
    for (int off = 16; off >= 1; off >>= 1) local += __shfl_xor(local, off, 32);
    if (lane == 0) sRed[wave] = local;
    __syncthreads();
    if (tid == 0) {
        float s = 0.0f;
        #pragma unroll
        for (int w = 0; w < 8; ++w) s += sRed[w];
        atomicAdd(loss, s * LOSS_SCALE);
    }
}

// ---------------------------------------------------------------------------
// K2: qf = quantized(32768x1024) @ out_w(1024x1024) + out_b.
// quantized is gathered on the fly from the L2-resident codebooks (never
// materialized in HBM).  Grid: (NROW/32, 8 col-groups of 128).  8 waves,
// each wave: one 16-col slice x TWO 16-row tiles -> each B fragment feeds
// two WMMAs; K accumulated in 4 chunks of 256 (chunk == group).
// ---------------------------------------------------------------------------
__global__ void __launch_bounds__(256)
pq_outgemm_kernel(const float* __restrict__ codebooks,
                  const int*   __restrict__ idx_ws,
                  const float* __restrict__ out_w,
                  const float* __restrict__ out_b,
                  float* __restrict__ qf) {
    using namespace pqcfg;
    __shared__ float sQ[ROWS2 * LDA];         // one 32x256 K-chunk of quantized
    __shared__ int   sIdx2[ROWS2 * Gdim];

    const int tid  = threadIdx.x;
    const int lane = tid & 31;
    const int wave = tid >> 5;
    const int n    = lane & 15;
    const int ksel = lane >> 4;
    const int r0   = blockIdx.x * ROWS2;
    const int col  = blockIdx.y * 128 + wave * 16 + n;

    if (tid < ROWS2 * Gdim)
        sIdx2[tid] = idx_ws[(size_t)(r0 + tid / Gdim) * Gdim + (tid % Gdim)];

    v8f acc0 = {}, acc1 = {};
    for (int gc = 0; gc < Gdim; ++gc) {       // K chunk == group
        __syncthreads();
        {   // gather quantized chunk: row m <- codebooks[gc][idx[m][gc]]
            const int m  = tid >> 3;          // 0..31
            const int e0 = tid & 7;
            const float* cq = codebooks + ((size_t)gc * Kdim + sIdx2[m * Gdim + gc]) * GD;
            #pragma unroll 4
            for (int j = 0; j < GD / 8; ++j) {
                const int e = e0 + j * 8;
                sQ[m * LDA + e] = cq[e];
            }
        }
        __syncthreads();
        const float* owg = out_w + (size_t)gc * GD * Edim;
        #pragma unroll 4
        for (int k0 = 0; k0 < GD; k0 += 4) {
            const int ka = k0 + 2 * ksel;
            v2f b;  b.x  = owg[(size_t)ka * Edim + col];
                    b.y  = owg[(size_t)(ka + 1) * Edim + col];
            v2f a0; a0.x = sQ[n * LDA + ka];        a0.y = sQ[n * LDA + ka + 1];
            v2f a1; a1.x = sQ[(16 + n) * LDA + ka]; a1.y = sQ[(16 + n) * LDA + ka + 1];
            acc0 = WMMA_F32(a0, b, acc0);
            acc1 = WMMA_F32(a1, b, acc1);
        }
    }
    const float bias = out_b[col];
    #pragma unroll
    for (int r = 0; r < 8; ++r) {
        const int m = r + 8 * ksel;
        __builtin_nontemporal_store(acc0[r] + bias,
                                    &qf[(size_t)(r0 + m) * Edim + col]);
        __builtin_nontemporal_store(acc1[r] + bias,
                                    &qf[(size_t)(r0 + 16 + m) * Edim + col]);
    }
}

// ---------------------------------------------------------------------------
extern "C" void kernel_launch(void* const* d_in, const int* in_sizes, int n_in,
                              void* d_out, int out_size, void* d_ws, size_t ws_size,
                              hipStream_t stream) {
    using namespace pqcfg;
    const float* features  = (const float*)d_in[0];
    const float* proj_w    = (const float*)d_in[1];
    const float* proj_b    = (const float*)d_in[2];
    const float* codebooks = (const float*)d_in[3];
    const float* out_w     = (const float*)d_in[4];
    const float* out_b     = (const float*)d_in[5];

    float* qf       = (float*)d_out;                       // [NROW*E]
    float* out_idx  = qf + (size_t)NROW * Edim;            // [NROW*G] (float-cast)
    float* loss     = out_idx + (size_t)NROW * Gdim;       // [1]

    float* c2ws   = (float*)d_ws;                          // [G*K]
    int*   idx_ws = (int*)((char*)d_ws + (size_t)Gdim * Kdim * sizeof(float));

    pq_prep_kernel<<<(Gdim * Kdim + 255) / 256, 256, 0, stream>>>(codebooks, c2ws, loss);

    dim3 g1(NROW / ROWS, Gdim);
    pq_quantize_kernel<<<g1, 256, 0, stream>>>(features, proj_w, proj_b, codebooks,
                                               c2ws, idx_ws, out_idx, loss);

    dim3 g2(NROW / ROWS2, Edim / 128);
    pq_outgemm_kernel<<<g2, 256, 0, stream>>>(codebooks, idx_ws, out_w, out_b, qf);
}